// VectorizedLSTM_56169582297154
// MI455X (gfx1250) — compile-verified
//
#include <hip/hip_runtime.h>

typedef __attribute__((ext_vector_type(2))) float v2f;
typedef __attribute__((ext_vector_type(8))) float v8f;

// Shapes (fixed by the reference): S=512, B=64, I=H=1024, 4H=4096.
#define SEQ   512
#define BATCH 64
#define HID   1024
#define BH    (BATCH * HID)   // 65536 elements per timestep slice

#define BK      32            // K-block per pipeline stage
#define ASTRIDE 36            // padded LDS row stride (floats): 36*ln mod 64 distinct
#define A_FL    (32 * ASTRIDE)            // 1152 floats per A buffer (32 rows x BK)
#define B_FL    (8 * 16 * ASTRIDE)        // 4608 floats per B buffer (8 tiles x 16 x BK)
#define BUF_FL  (A_FL + B_FL)             // 5760 floats per stage buffer

__device__ __forceinline__ float sigf(float x) { return 1.0f / (1.0f + __expf(-x)); }

// CDNA5 async global->LDS copy, 128 bits per lane (ASYNCcnt-tracked, no VGPR data).
__device__ __forceinline__ void async_b128(unsigned lds_byte_addr, const float* gptr) {
    asm volatile("global_load_async_to_lds_b128 %0, %1, off"
                 :: "v"(lds_byte_addr), "v"(gptr) : "memory");
}
__device__ __forceinline__ void wait_async0() {
    asm volatile("s_wait_asynccnt 0x0" ::: "memory");
}

// Issue the async copies for K-block kb into (Abuf, Bbuf).
// A: 32 rows (batch) x 32 k-cols; B: 8 N-tiles x 16 weight rows x 32 k-cols.
__device__ __forceinline__ void issue_block(
    int kb, float* Abuf, float* Bbuf,
    const float* __restrict__ xt, const float* __restrict__ hprev,
    const float* __restrict__ Wih, const float* __restrict__ Whh,
    int tid, int mbase, int wgj)
{
    const int k0 = kb << 5;
    const float* Asrc = (k0 < 1024) ? (xt + (size_t)mbase * 1024 + k0)
                                    : (hprev + (size_t)mbase * 1024 + (k0 - 1024));
    const float* Wsrc = (k0 < 1024) ? (Wih + k0) : (Whh + (k0 - 1024));

    {   // A: 256 float4 -> one per thread
        const int row = tid >> 3, c4 = tid & 7;
        async_b128((unsigned)(uintptr_t)&Abuf[row * ASTRIDE + (c4 << 2)],
                   Asrc + (size_t)row * 1024 + (c4 << 2));
    }
    #pragma unroll
    for (int i = 0; i < 4; ++i) {  // B: 1024 float4 -> four per thread
        const int f  = tid + (i << 8);
        const int tt = f >> 7, r = (f >> 3) & 15, c4 = f & 7;
        const int wr = ((tt >> 1) << 10) + (wgj << 5) + ((tt & 1) << 4) + r;
        async_b128((unsigned)(uintptr_t)&Bbuf[tt * (16 * ASTRIDE) + r * ASTRIDE + (c4 << 2)],
                   Wsrc + (size_t)wr * 1024 + (c4 << 2));
    }
}

// One LSTM timestep: gates = [x_t | h_prev] @ [W_ih | W_hh]^T + b, then pointwise.
// Grid (2, 32): blockIdx.x = batch half (M=32), blockIdx.y = 32-wide hidden-unit group.
// 8 waves; wave w owns gate (w>>1), n-half (w&1): one 16-wide N tile, two 16x16 M tiles.
__global__ __launch_bounds__(256) void lstm_step_kernel(
    const float* __restrict__ xt,     // (64,1024) slice of x at timestep t
    const float* __restrict__ hprev,  // (64,1024)
    const float* __restrict__ cin,    // (64,1024)
    const float* __restrict__ Wih,    // (4096,1024) row-major
    const float* __restrict__ Whh,    // (4096,1024) row-major
    const float* __restrict__ bih,    // (4096)
    const float* __restrict__ bhh,    // (4096)
    float* __restrict__ hout,         // (64,1024) -> d_out slice for t
    float* __restrict__ cout)         // (64,1024)
{
    __shared__ float smem[2 * BUF_FL];          // 46,080 bytes, double-buffered
    float* const A0 = smem;
    float* const B0 = smem + A_FL;
    float* const A1 = smem + BUF_FL;
    float* const B1 = smem + BUF_FL + A_FL;

    const int tid   = threadIdx.x;
    const int w     = tid >> 5;        // wave 0..7
    const int L     = tid & 31;        // lane
    const int ln    = L & 15;
    const int kb2   = (L >> 4) << 1;   // 0 or 2: K offset of this lane-half (frag layout)
    const int mbase = blockIdx.x << 5; // 0 or 32
    const int wgj   = blockIdx.y;      // 0..31
    const int gate  = w >> 1;          // 0..3 (i,f,g,o)
    const int half  = w & 1;
    const int nrow  = (gate << 10) + (wgj << 5) + (half << 4);

    const float bias_n = bih[nrow + ln] + bhh[nrow + ln];

    v8f acc0 = {};  // M rows mbase+0..15
    v8f acc1 = {};  // M rows mbase+16..31

    // Pipeline prologue: stage block 0.
    issue_block(0, A0, B0, xt, hprev, Wih, Whh, tid, mbase, wgj);
    wait_async0();
    __syncthreads();

    // K = 2048 total (x-part then h-part), 64 blocks of BK=32, 2-deep pipeline.
    for (int kb = 0; kb < 64; kb += 2) {
        // even stage: compute buf0, prefetch into buf1
        issue_block(kb + 1, A1, B1, xt, hprev, Wih, Whh, tid, mbase, wgj);
        {
            const float* Bt = B0 + w * (16 * ASTRIDE);
            #pragma unroll
            for (int k4 = 0; k4 < 8; ++k4) {
                const int kk = (k4 << 2) + kb2;
                const v2f bf = *(const v2f*)&Bt[ln * ASTRIDE + kk];
                const v2f a0 = *(const v2f*)&A0[ln * ASTRIDE + kk];
                const v2f a1 = *(const v2f*)&A0[(16 + ln) * ASTRIDE + kk];
                acc0 = __builtin_amdgcn_wmma_f32_16x16x4_f32(false, a0, false, bf,
                                                             (short)0, acc0, false, false);
                acc1 = __builtin_amdgcn_wmma_f32_16x16x4_f32(false, a1, false, bf,
                                                             (short)0, acc1, false, false);
            }
        }
        wait_async0();
        __syncthreads();

        // odd stage: compute buf1, prefetch into buf0
        if (kb + 2 < 64)
            issue_block(kb + 2, A0, B0, xt, hprev, Wih, Whh, tid, mbase, wgj);
        {
            const float* Bt = B1 + w * (16 * ASTRIDE);
            #pragma unroll
            for (int k4 = 0; k4 < 8; ++k4) {
                const int kk = (k4 << 2) + kb2;
                const v2f bf = *(const v2f*)&Bt[ln * ASTRIDE + kk];
                const v2f a0 = *(const v2f*)&A1[ln * ASTRIDE + kk];
                const v2f a1 = *(const v2f*)&A1[(16 + ln) * ASTRIDE + kk];
                acc0 = __builtin_amdgcn_wmma_f32_16x16x4_f32(false, a0, false, bf,
                                                             (short)0, acc0, false, false);
                acc1 = __builtin_amdgcn_wmma_f32_16x16x4_f32(false, a1, false, bf,
                                                             (short)0, acc1, false, false);
            }
        }
        wait_async0();
        __syncthreads();
    }

    // Stage gates (+bias) into LDS: gs[tile(=gate*2+half)][m 0..31][n 0..15].
    float* gs = smem;                  // 4096 floats, reuses stage-0 buffer
    const int mo = (L >> 4) << 3;      // C/D layout: lanes 16-31 hold M+8
    #pragma unroll
    for (int r = 0; r < 8; ++r) {
        gs[w * 512 + (mo + r) * 16 + ln]      = acc0[r] + bias_n;
        gs[w * 512 + (mo + r + 16) * 16 + ln] = acc1[r] + bias_n;
    }
    __syncthreads();

    // Pointwise LSTM update: 32 batches x 32 hidden units, 4 per thread.
    #pragma unroll
    for (int p0 = 0; p0 < 4; ++p0) {
        const int p  = tid + (p0 << 8);
        const int mb = p >> 5;
        const int jj = p & 31;
        const int hj = jj >> 4, nn = jj & 15;
        const int base = hj * 512 + mb * 16 + nn;   // gate stride = 1024 floats
        const float gi = gs[base];
        const float gf = gs[base + 1024];
        const float gg = gs[base + 2048];
        const float go = gs[base + 3072];

        const size_t idx = (size_t)(mbase + mb) * 1024 + (size_t)(wgj << 5) + jj;
        const float co = cin[idx];
        const float iv = sigf(gi);
        const float fv = sigf(gf);
        const float gv = tanhf(gg);
        const float ov = sigf(go);
        const float cn = fv * co + iv * gv;
        cout[idx] = cn;
        hout[idx] = ov * tanhf(cn);
    }
}

// Append final (h, c) after outputs: h_final == outputs[S-1], c from workspace.
__global__ __launch_bounds__(256) void lstm_tail_kernel(
    const float* __restrict__ hlast, const float* __restrict__ cbuf,
    float* __restrict__ tail)
{
    const int i = blockIdx.x * 256 + threadIdx.x;  // 0..65535
    tail[i]      = hlast[i];
    tail[BH + i] = cbuf[i];
}

extern "C" void kernel_launch(void* const* d_in, const int* in_sizes, int n_in,
                              void* d_out, int out_size, void* d_ws, size_t ws_size,
                              hipStream_t stream)
{
    (void)in_sizes; (void)n_in; (void)out_size; (void)ws_size;
    const float* x   = (const float*)d_in[0];  // (512,64,1024)
    const float* h0  = (const float*)d_in[1];  // (64,1024)
    const float* c0  = (const float*)d_in[2];  // (64,1024)
    const float* Wih = (const float*)d_in[3];  // (4096,1024)
    const float* Whh = (const float*)d_in[4];  // (4096,1024)
    const float* bih = (const float*)d_in[5];  // (4096)
    const float* bhh = (const float*)d_in[6];  // (4096)
    float* out  = (float*)d_out;               // outputs (512,64,1024) then h, c
    float* cbuf = (float*)d_ws;                // 64*1024 f32 = 256 KB scratch

    const dim3 grid(2, 32);
    for (int t = 0; t < SEQ; ++t) {
        const float* xt    = x + (size_t)t * BH;
        const float* hprev = (t == 0) ? h0 : (out + (size_t)(t - 1) * BH);
        const float* cin   = (t == 0) ? c0 : cbuf;
        float* hout        = out + (size_t)t * BH;
        lstm_step_kernel<<<grid, 256, 0, stream>>>(xt, hprev, cin, Wih, Whh,
                                                   bih, bhh, hout, cbuf);
    }
    lstm_tail_kernel<<<256, 256, 0, stream>>>(out + (size_t)(SEQ - 1) * BH, cbuf,
                                              out + (size_t)SEQ * BH);
}